// GCNLayer_5549097746957
// MI455X (gfx1250) — compile-verified
//
#include <hip/hip_runtime.h>
#include <hip/hip_bf16.h>

#define N_NODES   50000
#define N_EDGES   800000
#define IN_FEATS  128
#define OUT_FEATS 512
#define LEAKY     0.01f

typedef __attribute__((ext_vector_type(2))) float v2f;
typedef __attribute__((ext_vector_type(4))) float v4f;
typedef __attribute__((ext_vector_type(8))) float v8f;

// ---------------------------------------------------------------------------
// Kernel 1: zero the 50000x128 accumulator in workspace (float4 stores)
// ---------------------------------------------------------------------------
__global__ __launch_bounds__(256) void zero_h(float* __restrict__ h) {
    unsigned i = blockIdx.x * 256u + threadIdx.x;           // 1.6M float4 slots
    ((v4f*)h)[i] = (v4f){0.f, 0.f, 0.f, 0.f};
}

// ---------------------------------------------------------------------------
// Kernel 2: edge scatter-add. One wave32 per edge: lane l moves 4 consecutive
// floats (16B load) of feature[src] and atomically adds into h[dst].
// Atomics land in the 192MB L2 (accumulator is only 25.6MB).
// ---------------------------------------------------------------------------
__global__ __launch_bounds__(256) void scatter_add(const float* __restrict__ feature,
                                                   const int*   __restrict__ src,
                                                   const int*   __restrict__ dst,
                                                   float*       __restrict__ h) {
    unsigned gid  = blockIdx.x * 256u + threadIdx.x;
    unsigned e    = gid >> 5;        // edge index (one wave per edge)
    unsigned lane = gid & 31u;       // 32 lanes * 4 floats = 128 feats

    int s = src[e];
    int d = dst[e];

    const float* fp = feature + (size_t)s * IN_FEATS + lane * 4u;
    v4f val = *(const v4f*)fp;       // global_load_b128

    float* hp = h + (size_t)d * IN_FEATS + lane * 4u;
    atomicAdd(hp + 0, val.x);        // global_atomic_add_f32
    atomicAdd(hp + 1, val.y);
    atomicAdd(hp + 2, val.z);
    atomicAdd(hp + 3, val.w);
}

// ---------------------------------------------------------------------------
// Kernel 3: out = LeakyReLU(h @ W^T + b) with V_WMMA_F32_16X16X4_F32.
// Wave computes a 16(M) x 64(N) tile: A fragment amortized over 4 WMMA chains.
//   A (16x4 f32): lane L -> M = L%16, VGPR0/1 = K = 2*(L/16), 2*(L/16)+1
//   B (4x16 f32): lane L -> N = L%16, VGPR0/1 = K = 2*(L/16), 2*(L/16)+1
//   C/D (16x16)  : VGPR v -> M = v + 8*(L/16), N = L%16
// out[n,o] = sum_k h[n,k] * W[o,k]  =>  B[k,n] = W[obase+n, k]
// ---------------------------------------------------------------------------
__global__ __launch_bounds__(256) void gemm_wmma(const float* __restrict__ h,
                                                 const float* __restrict__ W,
                                                 const float* __restrict__ bias,
                                                 float*       __restrict__ out) {
    const unsigned waveInBlk = threadIdx.x >> 5;               // 8 waves / block
    const unsigned lane      = threadIdx.x & 31u;
    const unsigned tile      = blockIdx.x * 8u + waveInBlk;    // 25000 tiles total

    const unsigned rowTile  = tile >> 3;        // 3125 row tiles (50000/16)
    const unsigned colGroup = tile & 7u;        // 8 groups of 64 cols (512/64)

    const unsigned nbase = rowTile * 16u;
    const unsigned obase = colGroup * 64u;

    const unsigned M  = lane & 15u;             // row (A) / col (B) within tile
    const unsigned kg = lane >> 4;              // K-pair selector: 0 or 1

    v8f acc0 = {}, acc1 = {}, acc2 = {}, acc3 = {};

    const float* arow = h + (size_t)(nbase + M) * IN_FEATS;
    const float* b0   = W + (size_t)(obase + 0u  + M) * IN_FEATS;
    const float* b1   = W + (size_t)(obase + 16u + M) * IN_FEATS;
    const float* b2   = W + (size_t)(obase + 32u + M) * IN_FEATS;
    const float* b3   = W + (size_t)(obase + 48u + M) * IN_FEATS;

    #pragma unroll 4
    for (int kb = 0; kb < IN_FEATS; kb += 4) {
        const int ko = kb + 2 * (int)kg;
        v2f a  = *(const v2f*)(arow + ko);       // global_load_b64
        v2f f0 = *(const v2f*)(b0 + ko);
        v2f f1 = *(const v2f*)(b1 + ko);
        v2f f2 = *(const v2f*)(b2 + ko);
        v2f f3 = *(const v2f*)(b3 + ko);
        acc0 = __builtin_amdgcn_wmma_f32_16x16x4_f32(false, a, false, f0,
                                                     (short)0, acc0, false, false);
        acc1 = __builtin_amdgcn_wmma_f32_16x16x4_f32(false, a, false, f1,
                                                     (short)0, acc1, false, false);
        acc2 = __builtin_amdgcn_wmma_f32_16x16x4_f32(false, a, false, f2,
                                                     (short)0, acc2, false, false);
        acc3 = __builtin_amdgcn_wmma_f32_16x16x4_f32(false, a, false, f3,
                                                     (short)0, acc3, false, false);
    }

    // Epilogue: bias + LeakyReLU, scatter per C/D VGPR layout.
    v8f accs[4] = {acc0, acc1, acc2, acc3};
    #pragma unroll
    for (int t = 0; t < 4; ++t) {
        const unsigned ocol = obase + (unsigned)t * 16u + (lane & 15u);
        const float bv = bias[ocol];
        #pragma unroll
        for (int v = 0; v < 8; ++v) {
            const unsigned Mrow = (unsigned)v + 8u * kg;
            float val = accs[t][v] + bv;
            val = (val >= 0.f) ? val : LEAKY * val;
            out[(size_t)(nbase + Mrow) * OUT_FEATS + ocol] = val;
        }
    }
}

// ---------------------------------------------------------------------------
extern "C" void kernel_launch(void* const* d_in, const int* in_sizes, int n_in,
                              void* d_out, int out_size, void* d_ws, size_t ws_size,
                              hipStream_t stream) {
    const float* feature = (const float*)d_in[0];   // [50000,128]
    const int*   src     = (const int*)  d_in[1];   // [800000]
    const int*   dst     = (const int*)  d_in[2];   // [800000]
    const float* W       = (const float*)d_in[3];   // [512,128]
    const float* bias    = (const float*)d_in[4];   // [512]
    float*       out     = (float*)d_out;           // [50000,512]
    float*       h       = (float*)d_ws;            // [50000,128] accumulator

    // 1) zero accumulator: 6.4M floats = 1.6M float4 -> 6250 blocks exact
    zero_h<<<(N_NODES * IN_FEATS / 4) / 256, 256, 0, stream>>>(h);

    // 2) scatter-add: one wave per edge -> 800000*32/256 = 100000 blocks exact
    scatter_add<<<(N_EDGES * 32) / 256, 256, 0, stream>>>(feature, src, dst, h);

    // 3) WMMA GEMM + bias + LeakyReLU: 3125*8 = 25000 wave-tiles, 8 waves/block
    gemm_wmma<<<3125, 256, 0, stream>>>(h, W, bias, out);
}